// ConvSoftArgmax2d_5128190952122
// MI455X (gfx1250) — compile-verified
//
#include <hip/hip_runtime.h>
#include <cstdint>

namespace {

constexpr int H = 512, W = 512;          // input spatial dims
constexpr int HO = 256, WO = 256;        // output spatial dims (k=3,s=2,p=1)
constexpr int BC = 128;                  // fused batch*channels
constexpr int OTX = 64;                  // output tile width per block
constexpr int OTY = 16;                  // output tile height per block
constexpr int IROWS = 2 * OTY + 1;       // 33 staged input rows
constexpr int LSTR  = 2 * OTX + 8;       // 136 staged floats per row (16B-aligned cols)
constexpr int KLOADS = LSTR / 4;         // 34 x b128 async loads per row
constexpr float DEN_EPS = 1e-12f;
constexpr float T_EPS   = 1e-8f;

// CDNA5 async global->LDS copy (ASYNCcnt path). Dest operand is a VGPR holding
// the LDS byte offset; source is a 64-bit VGPR-pair global address (GV mode).
__device__ __forceinline__ void async_b128_to_lds(const float* g, uint32_t lds_off) {
  asm volatile("global_load_async_to_lds_b128 %0, %1, off"
               :: "v"(lds_off), "v"(g)
               : "memory");
}

__device__ __forceinline__ void wait_asynccnt0() {
  asm volatile("s_wait_asynccnt 0" ::: "memory");
}

__global__ __launch_bounds__(256) void conv_softargmax2d(
    const float* __restrict__ x,      // (BC, H, W)
    const float* __restrict__ temp,   // scalar
    float* __restrict__ out)          // coords (BC,2,HO,WO) then resp (BC,HO,WO)
{
  __shared__ float tile[IROWS * LSTR];

  const int tid = threadIdx.x;
  const int ox0 = blockIdx.x * OTX;
  const int oy0 = blockIdx.y * OTY;
  const int c   = blockIdx.z;

  const int ry0 = 2 * oy0 - 1;   // first input row needed by this tile
  const int cb  = 2 * ox0 - 4;   // first staged input col (multiple of 4 -> 16B aligned)

  const float* xc = x + (size_t)c * (H * W);
  const uint32_t lds_base = (uint32_t)(uintptr_t)(&tile[0]);

  // ---- Stage input tile into LDS via async tensor-path loads -------------
  for (int l = tid; l < IROWS * KLOADS; l += 256) {
    const int row = l / KLOADS;
    const int k   = l - row * KLOADS;
    const int iy  = ry0 + row;
    const int gx4 = cb + 4 * k;                       // 4-float column group
    if ((unsigned)iy < (unsigned)H && (unsigned)gx4 < (unsigned)W) {
      async_b128_to_lds(xc + (size_t)iy * W + gx4,
                        lds_base + (uint32_t)(row * LSTR + 4 * k) * 4u);
    }
    // out-of-range quads skipped: compute masks those taps to e = 0
  }
  wait_asynccnt0();
  __syncthreads();

  // ---- Compute -----------------------------------------------------------
  const float invT = 1.0f / fmaxf(temp[0], T_EPS);
  const float nrm  = 2.0f / (float)(W - 1);           // H == W

  const int tx  = tid & (OTX - 1);                    // 0..63
  const int tyq = tid >> 6;                           // 0..3

  const size_t plane = (size_t)HO * WO;
  float* coords = out;
  float* resp   = out + (size_t)BC * 2 * plane;

#pragma unroll
  for (int q = 0; q < OTY / 4; ++q) {
    const int ty = tyq * (OTY / 4) + q;               // 0..15
    const int oy = oy0 + ty;
    const int ox = ox0 + tx;

    float s = 0.f, sv = 0.f, sx = 0.f;
    float rowsum[3];
#pragma unroll
    for (int i = 0; i < 3; ++i) {
      const int  iy  = 2 * oy - 1 + i;
      const bool rok = (unsigned)iy < (unsigned)H;    // only fails at oy==0, i==0
      const float* rp = &tile[(2 * ty + i) * LSTR + (2 * tx + 3)];
      const int ixm1 = 2 * ox - 1;                    // leftmost tap column
      const float v0 = rp[0], v1 = rp[1], v2 = rp[2];
      // zero-padded x_exp semantics: out-of-range taps contribute e = 0
      const float e0 = (rok && (unsigned)ixm1 < (unsigned)W) ? __expf(v0 * invT) : 0.f;
      const float e1 = rok ? __expf(v1 * invT) : 0.f;  // col 2*ox always in range
      const float e2 = rok ? __expf(v2 * invT) : 0.f;  // col 2*ox+1 <= 511 always
      const float rs = e0 + e1 + e2;
      rowsum[i] = rs;
      s  += rs;
      sv += e0 * v0 + e1 * v1 + e2 * v2;
      sx += e2 - e0;                                   // gx = [-1, 0, +1]
    }
    const float sy   = rowsum[2] - rowsum[0];          // gy = [-1, 0, +1]
    const float rden = 1.0f / (s + DEN_EPS);

    const float cxp = sx * rden + (float)(2 * ox);     // + center-pixel grid x
    const float cyp = sy * rden + (float)(2 * oy);     // + center-pixel grid y

    const size_t pix = (size_t)oy * WO + ox;
    float* cc = coords + (size_t)c * 2 * plane;
    cc[pix]          = cxp * nrm - 1.0f;               // normalized x
    cc[plane + pix]  = cyp * nrm - 1.0f;               // normalized y
    resp[(size_t)c * plane + pix] = sv * rden;         // softmax-pooled value
  }
}

} // namespace

extern "C" void kernel_launch(void* const* d_in, const int* in_sizes, int n_in,
                              void* d_out, int out_size, void* d_ws, size_t ws_size,
                              hipStream_t stream) {
  (void)in_sizes; (void)n_in; (void)out_size; (void)d_ws; (void)ws_size;
  const float* x = (const float*)d_in[0];
  const float* t = (const float*)d_in[1];
  float* out = (float*)d_out;

  dim3 grid(WO / OTX, HO / OTY, BC);   // (4, 16, 128)
  conv_softargmax2d<<<grid, dim3(256), 0, stream>>>(x, t, out);
}